// CompetitiveGNNModel_41111426957444
// MI455X (gfx1250) — compile-verified
//
#include <hip/hip_runtime.h>
#include <hip/hip_bf16.h>
#include <math.h>

typedef __attribute__((ext_vector_type(2))) float v2f;
typedef __attribute__((ext_vector_type(4))) float v4f;
typedef __attribute__((ext_vector_type(8))) float v8f;

#define N_NODES  100000
#define N_EDGES  3200000
#define F_IN     128
#define HID      64
#define N_GRAPHS 256

// ---------------- degree / normalization ----------------
__global__ void init_deg_kernel(float* deg, int n) {
    int i = blockIdx.x * blockDim.x + threadIdx.x;
    if (i < n) deg[i] = 1.0f;  // self-loop contributes 1
}

__global__ void count_deg_kernel(const int* __restrict__ dst, float* deg, int nE) {
    int e = blockIdx.x * blockDim.x + threadIdx.x;
    if (e < nE) atomicAdd(&deg[dst[e]], 1.0f);
}

__global__ void dinv_kernel(float* deg, int n) {
    int i = blockIdx.x * blockDim.x + threadIdx.x;
    if (i < n) {
        float d = deg[i];
        deg[i] = (d > 0.0f) ? rsqrtf(d) : 0.0f;   // in-place deg -> deg^{-1/2}
    }
}

// ---------------- dense transform: C[M,N] = (relu?)A[M,K] @ W[K,N] ----------------
// One wave computes one 16x16 tile with V_WMMA_F32_16X16X4_F32, K stepped by 4.
// f32 A 16x4 layout (ISA 7.12.2): lane l -> row m=l&15, khalf=l>>4;
//   VGPR0 = A[m][k0+2*khalf], VGPR1 = A[m][k0+2*khalf+1].
// f32 B 4x16 mirrors it on columns. C/D: VGPR r -> row r+8*(l>>4), col l&15.
// RELU is a compile-time flag so the K-loop stays branch/cndmask-free.
template <int RELU>
__global__ __launch_bounds__(256) void gemm_wmma_f32_kernel(
    const float* __restrict__ A, const float* __restrict__ W,
    float* __restrict__ C, int M, int K, int N)
{
    const int wave = threadIdx.x >> 5;
    const int lane = threadIdx.x & 31;
    const int mtile = blockIdx.x * 2 + (wave >> 2);   // 8 waves: 2 M-tiles x 4 N-tiles
    const int ntile = wave & 3;                        // N == 64 -> 4 tiles
    const int row0 = mtile * 16;
    const int col0 = ntile * 16;
    if (row0 >= M) return;

    const int mrow = lane & 15;
    const int kh   = lane >> 4;        // 0 or 1

    v8f acc = {};
    const float* Arow = A + (size_t)(row0 + mrow) * (size_t)K;
    const int bn = col0 + (lane & 15);

    for (int k0 = 0; k0 < K; k0 += 4) {
        const int ka = k0 + 2 * kh;
        v2f a = *(const v2f*)(Arow + ka);               // single global_load_b64
        if (RELU) {
            a.x = fmaxf(a.x, 0.0f);
            a.y = fmaxf(a.y, 0.0f);
        }
        v2f b;
        b.x = W[(size_t)ka * N + bn];
        b.y = W[(size_t)(ka + 1) * N + bn];
        acc = __builtin_amdgcn_wmma_f32_16x16x4_f32(
            /*neg_a=*/false, a, /*neg_b=*/false, b,
            /*c_mod=*/(short)0, acc, /*reuse_a=*/false, /*reuse_b=*/false);
    }

    float* Cout = C + (size_t)row0 * N + col0;
    #pragma unroll
    for (int r = 0; r < 8; ++r) {
        int m = r + 8 * kh;
        Cout[(size_t)m * N + (lane & 15)] = acc[r];
    }
}

// ---------------- GCN aggregation ----------------
// out[i][f] = b[f] + h[i][f] * dinv[i]^2  (self-loop + bias; initializes out)
// One thread per 4 features: b128 load + b128 store.
__global__ void selfloop_bias_kernel(const float* __restrict__ h,
                                     const float* __restrict__ dinv,
                                     const float* __restrict__ b,
                                     float* __restrict__ out, int n) {
    int idx = blockIdx.x * blockDim.x + threadIdx.x;   // n * HID/4 threads
    if (idx >= n * (HID / 4)) return;
    int node = idx >> 4;          // HID/4 == 16 quads per node
    int q    = (idx & 15) * 4;
    float w  = dinv[node];
    float w2 = w * w;
    v4f hv = *(const v4f*)(h + (size_t)node * HID + q);
    v4f bv = *(const v4f*)(b + q);
    v4f o;
    o.x = bv.x + hv.x * w2;
    o.y = bv.y + hv.y * w2;
    o.z = bv.z + hv.z * w2;
    o.w = bv.w + hv.w * w2;
    *(v4f*)(out + (size_t)node * HID + q) = o;
}

// One wave per edge: lane handles the contiguous feature pair {2*lane, 2*lane+1}
// -> one global_load_b64 gather + two global_atomic_add_f32 per lane.
__global__ __launch_bounds__(256) void edge_scatter_kernel(
    const float* __restrict__ h, const float* __restrict__ dinv,
    const int* __restrict__ src, const int* __restrict__ dst,
    float* __restrict__ out, int nE)
{
    long long tid = (long long)blockIdx.x * blockDim.x + threadIdx.x;
    long long e = tid >> 5;
    if (e >= nE) return;
    int lane = (int)(tid & 31);
    int s = src[e], d = dst[e];
    float w = dinv[s] * dinv[d];
    const float* hs = h + (size_t)s * HID + 2 * lane;
    float*       od = out + (size_t)d * HID + 2 * lane;
    v2f hv = *(const v2f*)hs;
    atomicAdd(&od[0], hv.x * w);
    atomicAdd(&od[1], hv.y * w);
}

// ---------------- pooling ----------------
__global__ void zero_kernel(float* p, int n) {
    int i = blockIdx.x * blockDim.x + threadIdx.x;
    if (i < n) p[i] = 0.0f;
}

__global__ void pool_sum_kernel(const float* __restrict__ h, const int* __restrict__ batch,
                                float* __restrict__ sums, float* __restrict__ cnt, int n) {
    int idx = blockIdx.x * blockDim.x + threadIdx.x;
    if (idx >= n * HID) return;
    int node = idx >> 6;
    int f    = idx & 63;
    int g    = batch[node];
    atomicAdd(&sums[(size_t)g * HID + f], h[idx]);
    if (f == 0) atomicAdd(&cnt[g], 1.0f);
}

__global__ void pool_mean_kernel(float* sums, const float* cnt, int ng) {
    int idx = blockIdx.x * blockDim.x + threadIdx.x;
    if (idx < ng * HID) {
        int g = idx >> 6;
        sums[idx] /= fmaxf(cnt[g], 1.0f);
    }
}

// ---------------- classifier MLP ----------------
__global__ void mlp1_kernel(const float* __restrict__ g, const float* __restrict__ Wc1,
                            const float* __restrict__ bc1, float* __restrict__ z) {
    int idx = blockIdx.x * blockDim.x + threadIdx.x;   // N_GRAPHS*32 threads
    if (idx >= N_GRAPHS * 32) return;
    int gi = idx >> 5;
    int j  = idx & 31;
    float acc = bc1[j];
    const float* gr = g + (size_t)gi * HID;
    #pragma unroll
    for (int k = 0; k < HID; ++k) acc += gr[k] * Wc1[k * 32 + j];
    z[idx] = fmaxf(acc, 0.0f);
}

__global__ void mlp2_kernel(const float* __restrict__ z, const float* __restrict__ Wc2,
                            const float* __restrict__ bc2, float* __restrict__ out) {
    int gi = blockIdx.x * blockDim.x + threadIdx.x;
    if (gi >= N_GRAPHS) return;
    float acc = bc2[0];
    const float* zr = z + (size_t)gi * 32;
    #pragma unroll
    for (int k = 0; k < 32; ++k) acc += zr[k] * Wc2[k];
    out[gi] = 1.0f / (1.0f + expf(-acc));
}

// ---------------- orchestration ----------------
extern "C" void kernel_launch(void* const* d_in, const int* in_sizes, int n_in,
                              void* d_out, int out_size, void* d_ws, size_t ws_size,
                              hipStream_t stream) {
    (void)in_sizes; (void)n_in; (void)out_size; (void)ws_size;

    const float* x     = (const float*)d_in[0];
    const int*   eidx  = (const int*)  d_in[1];
    const int*   batch = (const int*)  d_in[2];
    const float* W1    = (const float*)d_in[3];
    const float* b1    = (const float*)d_in[4];
    const float* W2    = (const float*)d_in[5];
    const float* b2    = (const float*)d_in[6];
    const float* W3    = (const float*)d_in[7];
    const float* b3    = (const float*)d_in[8];
    const float* Wc1   = (const float*)d_in[9];
    const float* bc1   = (const float*)d_in[10];
    const float* Wc2   = (const float*)d_in[11];
    const float* bc2   = (const float*)d_in[12];

    const int* src = eidx;              // edge_index[0]
    const int* dst = eidx + N_EDGES;    // edge_index[1]

    // workspace layout (floats)
    float* bufA = (float*)d_ws;                       // [N_NODES*HID] gemm out
    float* bufB = bufA + (size_t)N_NODES * HID;       // [N_NODES*HID] aggregated
    float* dinv = bufB + (size_t)N_NODES * HID;       // [N_NODES] deg -> deg^-1/2
    float* sums = dinv + N_NODES;                     // [N_GRAPHS*HID]
    float* cnt  = sums + (size_t)N_GRAPHS * HID;      // [N_GRAPHS]
    float* z    = cnt + N_GRAPHS;                     // [N_GRAPHS*32]

    const int B = 256;
    const int nodeBlocks = (N_NODES + B - 1) / B;
    const int edgeBlocks = (N_EDGES + B - 1) / B;
    const int featBlocks = (N_NODES * HID + B - 1) / B;
    const int quadBlocks = (N_NODES * (HID / 4) + B - 1) / B;
    const long long scatterThreads = (long long)N_EDGES * 32;
    const int scatterBlocks = (int)((scatterThreads + B - 1) / B);
    const int gemmBlocks = (N_NODES / 16) / 2;        // 2 M-tiles per block

    // degree -> dinv (graph is static across the three layers)
    init_deg_kernel<<<nodeBlocks, B, 0, stream>>>(dinv, N_NODES);
    count_deg_kernel<<<edgeBlocks, B, 0, stream>>>(dst, dinv, N_EDGES);
    dinv_kernel<<<nodeBlocks, B, 0, stream>>>(dinv, N_NODES);

    // ---- layer 1: h1 = x @ W1 ; agg1 = b1 + norm-aggregate(h1) ----
    gemm_wmma_f32_kernel<0><<<gemmBlocks, B, 0, stream>>>(x, W1, bufA, N_NODES, F_IN, HID);
    selfloop_bias_kernel<<<quadBlocks, B, 0, stream>>>(bufA, dinv, b1, bufB, N_NODES);
    edge_scatter_kernel<<<scatterBlocks, B, 0, stream>>>(bufA, dinv, src, dst, bufB, N_EDGES);

    // ---- layer 2: h2 = relu(agg1) @ W2 ; agg2 = b2 + aggregate(h2) ----
    gemm_wmma_f32_kernel<1><<<gemmBlocks, B, 0, stream>>>(bufB, W2, bufA, N_NODES, HID, HID);
    selfloop_bias_kernel<<<quadBlocks, B, 0, stream>>>(bufA, dinv, b2, bufB, N_NODES);
    edge_scatter_kernel<<<scatterBlocks, B, 0, stream>>>(bufA, dinv, src, dst, bufB, N_EDGES);

    // ---- layer 3: h3 = relu(agg2) @ W3 ; agg3 = b3 + aggregate(h3) (no relu after) ----
    gemm_wmma_f32_kernel<1><<<gemmBlocks, B, 0, stream>>>(bufB, W3, bufA, N_NODES, HID, HID);
    selfloop_bias_kernel<<<quadBlocks, B, 0, stream>>>(bufA, dinv, b3, bufB, N_NODES);
    edge_scatter_kernel<<<scatterBlocks, B, 0, stream>>>(bufA, dinv, src, dst, bufB, N_EDGES);

    // ---- global mean pool over batch ----
    const int poolZero = N_GRAPHS * HID + N_GRAPHS;   // sums and cnt are contiguous
    zero_kernel<<<(poolZero + B - 1) / B, B, 0, stream>>>(sums, poolZero);
    pool_sum_kernel<<<featBlocks, B, 0, stream>>>(bufB, batch, sums, cnt, N_NODES);
    pool_mean_kernel<<<(N_GRAPHS * HID + B - 1) / B, B, 0, stream>>>(sums, cnt, N_GRAPHS);

    // ---- classifier ----
    mlp1_kernel<<<(N_GRAPHS * 32 + B - 1) / B, B, 0, stream>>>(sums, Wc1, bc1, z);
    mlp2_kernel<<<(N_GRAPHS + B - 1) / B, B, 0, stream>>>(z, Wc2, bc2, (float*)d_out);
}